// GCNLayer_10780367913772
// MI455X (gfx1250) — compile-verified
//
#include <hip/hip_runtime.h>

// SpMM (COO): out[row[e]] += val[e] * embeds[col[e]]   E=3.2M, D=128, N=100k
// Memory-bound: embeds (51MB) + out (51MB) both L2-resident (192MB L2); edge
// stream (38MB) is double-buffered into LDS via CDNA5 async global->LDS copies
// (ASYNCcnt), hiding its latency behind the gather/atomic inner loop.

#define D_FEAT 128
#define WPB 8           // waves per block (256 threads / wave32)
#define GRP 32          // edges per group (one wave-iteration)

__global__ __launch_bounds__(256) void gcn_zero_kernel(float4* __restrict__ out, int n4) {
    int i = blockIdx.x * blockDim.x + threadIdx.x;
    if (i < n4) out[i] = make_float4(0.f, 0.f, 0.f, 0.f);
}

// CDNA5 async copy: one b32 per lane, global -> LDS, tracked by ASYNCcnt.
__device__ __forceinline__ void async_g2l_b32(unsigned lds_addr, const void* gptr) {
    asm volatile("global_load_async_to_lds_b32 %0, %1, off"
                 :: "v"(lds_addr), "v"((unsigned long long)(uintptr_t)gptr)
                 : "memory");
}

__global__ __launch_bounds__(256) void gcn_spmm_kernel(
    const int*   __restrict__ edge_row,
    const int*   __restrict__ edge_col,
    const float* __restrict__ edge_val,
    const float* __restrict__ embeds,
    float*       __restrict__ out,
    int n_edges)
{
    // Per-wave double buffer: [rows 0..31 | cols 32..63 | val-bits 64..95]
    __shared__ int smem[WPB][2][3 * GRP];

    const int lane    = threadIdx.x & 31;
    const int wv      = threadIdx.x >> 5;
    const int wave    = (int)((blockIdx.x * blockDim.x + threadIdx.x) >> 5);
    const int n_waves = (int)((gridDim.x * blockDim.x) >> 5);
    const int stride  = n_waves * GRP;

    int base = wave * GRP;
    if (base >= n_edges) return;

    // LDS byte offsets (generic ptr truncated to addr[31:0] = LDS offset).
    const unsigned lds0 = (unsigned)(uintptr_t)&smem[wv][0][0];
    const unsigned lds1 = (unsigned)(uintptr_t)&smem[wv][1][0];

    // Prologue: issue async copies for group 0 into buffer 0 (3 x ASYNCcnt).
    {
        const int e = min(base + lane, n_edges - 1);   // clamp tail (dup, unused)
        async_g2l_b32(lds0 + 4u * (unsigned)lane,             edge_row + e);
        async_g2l_b32(lds0 + 4u * (unsigned)(GRP + lane),     edge_col + e);
        async_g2l_b32(lds0 + 4u * (unsigned)(2 * GRP + lane), edge_val + e);
    }

    int buf = 0;
    for (; base < n_edges; base += stride, buf ^= 1) {
        const int  nb       = base + stride;
        const bool has_next = nb < n_edges;            // wave-uniform

        // Issue next group's async copies into the other buffer.
        if (has_next) {
            const unsigned l = buf ? lds0 : lds1;
            const int e = min(nb + lane, n_edges - 1);
            async_g2l_b32(l + 4u * (unsigned)lane,             edge_row + e);
            async_g2l_b32(l + 4u * (unsigned)(GRP + lane),     edge_col + e);
            async_g2l_b32(l + 4u * (unsigned)(2 * GRP + lane), edge_val + e);
        }

        // Retire exactly the current buffer's 3 copies (in-order completion);
        // the 3 just-issued copies may remain in flight.
        if (has_next) asm volatile("s_wait_asynccnt 0x3" ::: "memory");
        else          asm volatile("s_wait_asynccnt 0x0" ::: "memory");

        const int* sb  = &smem[wv][buf][0];
        const int  cnt = min(GRP, n_edges - base);

        // CDNA5 prefetch: each lane pulls its own edge's 512B embedding row
        // (4 x 128B lines) from L2 into the WGP cache ahead of the serial loop.
        {
            const int c_mine = sb[GRP + lane];
            const float* rowp = embeds + (size_t)c_mine * D_FEAT;
            __builtin_prefetch(rowp,      0, 3);
            __builtin_prefetch(rowp + 32, 0, 3);
            __builtin_prefetch(rowp + 64, 0, 3);
            __builtin_prefetch(rowp + 96, 0, 3);
        }

        for (int j = 0; j < cnt; ++j) {
            // Uniform-address LDS reads -> hardware broadcast to all lanes.
            const int   r = sb[j];
            const int   c = sb[GRP + j];
            const float v = __int_as_float(sb[2 * GRP + j]);

            // Coalesced 512B gather: lane l loads floats [4l,4l+4) of row c.
            const float4 x = ((const float4*)(embeds + (size_t)c * D_FEAT))[lane];

            // Coalesced no-return global_atomic_add_f32 into out[r] (L2-resident).
            float* o = out + (size_t)r * D_FEAT + lane * 4;
            atomicAdd(o + 0, x.x * v);
            atomicAdd(o + 1, x.y * v);
            atomicAdd(o + 2, x.z * v);
            atomicAdd(o + 3, x.w * v);
        }
    }
}

extern "C" void kernel_launch(void* const* d_in, const int* in_sizes, int n_in,
                              void* d_out, int out_size, void* d_ws, size_t ws_size,
                              hipStream_t stream) {
    const int*   edge_row = (const int*)  d_in[0];
    const int*   edge_col = (const int*)  d_in[1];
    const float* edge_val = (const float*)d_in[2];
    const float* embeds   = (const float*)d_in[3];
    float*       out      = (float*)d_out;

    const int n_edges = in_sizes[0];

    // 1) Zero the output (atomics accumulate; harness doesn't re-poison).
    const int n4 = out_size / 4;
    gcn_zero_kernel<<<(n4 + 255) / 256, 256, 0, stream>>>((float4*)out, n4);

    // 2) SpMM: one wave per 32-edge group (100000 groups, 8 waves/block).
    const int groups = (n_edges + GRP - 1) / GRP;
    const int blocks = (groups + WPB - 1) / WPB;
    gcn_spmm_kernel<<<blocks, 256, 0, stream>>>(edge_row, edge_col, edge_val,
                                                embeds, out, n_edges);
}